// SPConv2Dkxk_71536975282579
// MI455X (gfx1250) — compile-verified
//
#include <hip/hip_runtime.h>

typedef __attribute__((ext_vector_type(2))) float    v2f;
typedef __attribute__((ext_vector_type(8))) float    v8f;
typedef __attribute__((ext_vector_type(2))) unsigned v2u;

#define C1V    64
#define C2V    64
#define KKV    576           // C1*3*3, GEMM K dimension
#define LDSROW 578           // padded LDS row stride (dwords): conflict-free, b64-aligned
#define WOFF   (C2V * LDSROW)   // dword offset of tab[] after W in LDS

// out[N,64] = gather_patches(x, idx)[N,576] @ W^T[576,64] + bias
// One wave computes a 16(N-rows) x 64(C2) tile via 4x V_WMMA_F32_16X16X4_F32 accumulators.
__global__ __launch_bounds__(256)
void SPConv2Dkxk_wmma_f32_kernel(const float* __restrict__ x,
                                 const int*   __restrict__ idx,
                                 const float* __restrict__ w,
                                 const float* __restrict__ bias,
                                 float*       __restrict__ out,
                                 int ntiles)
{
    extern __shared__ float smem[];                     // [64][578] W + [576] tab
    unsigned* tab = (unsigned*)(smem + WOFF);

    // ---- stage W into padded LDS ----
#pragma unroll 4
    for (int i = threadIdx.x; i < C2V * KKV; i += 256) {
        int c2 = i / KKV;
        int k  = i - c2 * KKV;
        smem[c2 * LDSROW + k] = w[i];
    }
    // ---- stage packed k -> (rel-offset, r) table ----
    for (int k = threadIdx.x; k < KKV; k += 256) {
        int c  = k / 9;
        int r  = k - 9 * c;
        int dy = (r * 11) >> 5;                         // r/3 for r in 0..8
        int dx = r - 3 * dy;
        int off_rel = (c << 16) + (dy - 1) * 256 + (dx - 1);
        tab[k] = (unsigned)(off_rel + 257) | ((unsigned)r << 22);
    }
    __syncthreads();

    const int lane = threadIdx.x & 31;
    const int wave = threadIdx.x >> 5;
    const int m    = lane & 15;      // row-within-tile (A) / N-within-tile (B,D)
    const int g    = lane >> 4;      // K-group for A/B, M-group for D

    const int tile = blockIdx.x * 8 + wave;
    if (tile >= ntiles) return;      // wave-uniform: EXEC stays all-ones for WMMA
    const int row_base = tile * 16;
    const int row      = row_base + m;

    const int bi = idx[3 * row + 0];
    const int yi = idx[3 * row + 1];
    const int xi = idx[3 * row + 2];

    // 32-bit element base: center element minus the +257 packing bias
    const unsigned pbase = ((unsigned)bi << 22) + ((unsigned)yi << 8) + (unsigned)xi - 257u;

    // one-time 9-bit validity mask: bit r -> (yi+dy-1, xi+dx-1) inside [0,256)
    const int y0 = yi - 1, x0 = xi - 1;
    unsigned ymask = 0, xmask = 0;
#pragma unroll
    for (int d = 0; d < 3; ++d) {
        ymask |= ((unsigned)(y0 + d) < 256u ? 1u : 0u) << d;
        xmask |= ((unsigned)(x0 + d) < 256u ? 1u : 0u) << d;
    }
    // vmask bit (dy*3+dx) = ymask[dy] & xmask[dx]
    unsigned vmask = 0;
#pragma unroll
    for (int dyi = 0; dyi < 3; ++dyi)
        vmask |= ((ymask >> dyi) & 1u ? (xmask & 7u) : 0u) << (3 * dyi);

    v8f acc0 = {}, acc1 = {}, acc2 = {}, acc3 = {};

#pragma unroll 4
    for (int k0 = 0; k0 < KKV; k0 += 4) {
        const int kb = k0 + 2 * g;                      // even -> 8B-aligned pairs

        // ---- A fragment: two consecutive K via one b64 table load ----
        v2u tp = *(const v2u*)&tab[kb];

        unsigned t0  = tp.x;
        unsigned r0  = t0 >> 22;
        bool     ok0 = (vmask >> r0) & 1u;
        unsigned e0  = pbase + (ok0 ? (t0 & 0x3FFFFFu) : 257u);
        float    v0  = x[(size_t)e0];
        float    a0  = ok0 ? v0 : 0.0f;

        unsigned t1  = tp.y;
        unsigned r1  = t1 >> 22;
        bool     ok1 = (vmask >> r1) & 1u;
        unsigned e1  = pbase + (ok1 ? (t1 & 0x3FFFFFu) : 257u);
        float    v1  = x[(size_t)e1];
        float    a1  = ok1 ? v1 : 0.0f;

        v2f A; A.x = a0; A.y = a1;

        // ---- B fragments from LDS: B[K][n] = W[n][K]; one ds_load_b64 per N-tile ----
        v2f B0 = *(const v2f*)&smem[( 0 + m) * LDSROW + kb];
        v2f B1 = *(const v2f*)&smem[(16 + m) * LDSROW + kb];
        v2f B2 = *(const v2f*)&smem[(32 + m) * LDSROW + kb];
        v2f B3 = *(const v2f*)&smem[(48 + m) * LDSROW + kb];

        acc0 = __builtin_amdgcn_wmma_f32_16x16x4_f32(false, A, false, B0, (short)0, acc0, false, false);
        acc1 = __builtin_amdgcn_wmma_f32_16x16x4_f32(false, A, false, B1, (short)0, acc1, false, false);
        acc2 = __builtin_amdgcn_wmma_f32_16x16x4_f32(false, A, false, B2, (short)0, acc2, false, false);
        acc3 = __builtin_amdgcn_wmma_f32_16x16x4_f32(false, A, false, B3, (short)0, acc3, false, false);
    }

    // ---- epilogue: D layout = VGPR v holds M = v + 8g, N = t*16 + m ----
    float* op = out + (size_t)(row_base + 8 * g) * C2V + m;
    {
        float bv = bias[ 0 + m];
#pragma unroll
        for (int v = 0; v < 8; ++v) op[v * C2V +  0] = acc0[v] + bv;
    }
    {
        float bv = bias[16 + m];
#pragma unroll
        for (int v = 0; v < 8; ++v) op[v * C2V + 16] = acc1[v] + bv;
    }
    {
        float bv = bias[32 + m];
#pragma unroll
        for (int v = 0; v < 8; ++v) op[v * C2V + 32] = acc2[v] + bv;
    }
    {
        float bv = bias[48 + m];
#pragma unroll
        for (int v = 0; v < 8; ++v) op[v * C2V + 48] = acc3[v] + bv;
    }
}

extern "C" void kernel_launch(void* const* d_in, const int* in_sizes, int n_in,
                              void* d_out, int out_size, void* d_ws, size_t ws_size,
                              hipStream_t stream) {
    const float* x    = (const float*)d_in[0];
    const int*   idx  = (const int*)  d_in[1];
    const float* w    = (const float*)d_in[2];
    const float* bias = (const float*)d_in[3];
    float* out = (float*)d_out;

    const int N      = in_sizes[1] / 3;      // 131072
    const int ntiles = N / 16;               // 8192
    const int blocks = (ntiles + 7) / 8;     // 8 waves (tiles) per 256-thread block

    const size_t lds_bytes = (size_t)(WOFF + KKV) * sizeof(float);  // 150272 B < 320 KB
    hipFuncSetAttribute((const void*)SPConv2Dkxk_wmma_f32_kernel,
                        hipFuncAttributeMaxDynamicSharedMemorySize, (int)lds_bytes);

    SPConv2Dkxk_wmma_f32_kernel<<<blocks, 256, lds_bytes, stream>>>(x, idx, w, bias, out, ntiles);
}